// FrequencyModulationLayer_72241349918967
// MI455X (gfx1250) — compile-verified
//
#include <hip/hip_runtime.h>
#include <math.h>

// ---------------------------------------------------------------------------
// FrequencyModulationLayer on MI455X (gfx1250, wave32).
//   y = irfft( smooth(|rfft(pad(x)) * gain|) * e^{i*phase} )[:L]
// 256 signals, L=65537, NFFT=131072. Real FFT via 65536-pt complex FFT of
// packed even/odd samples; 65536 = 256*256 four-step; each 256-pt FFT = two
// radix-16 stages, each stage = complex 16x16 matmul on V_WMMA_F32_16X16X4_F32
// (full fp32 precision; problem is HBM-bandwidth bound, ~1.1GB total traffic).
// Data movement uses CDNA5 async global<->LDS ops (ASYNCcnt) where available.
// Workspace: gain (512KB region) | bufA (S*65536 float2) | bufB (same).
// Requires ws_size >= 0.5MB + 2*S*512KB (S=256 -> ~257MB).
// ---------------------------------------------------------------------------

#define PI_F 3.14159265358979323846f

typedef __attribute__((ext_vector_type(2))) float v2f;
typedef __attribute__((ext_vector_type(8))) float v8f;

typedef int i32x4 __attribute__((vector_size(16)));
typedef int i32x2 __attribute__((vector_size(8)));

#define AS1 __attribute__((address_space(1)))
#define AS3 __attribute__((address_space(3)))

#if defined(__has_builtin)
#if __has_builtin(__builtin_amdgcn_global_load_async_to_lds_b128) &&   \
    __has_builtin(__builtin_amdgcn_global_load_async_to_lds_b64) &&    \
    __has_builtin(__builtin_amdgcn_global_store_async_from_lds_b64) && \
    __has_builtin(__builtin_amdgcn_global_store_async_from_lds_b32)
#define HAVE_ASYNC_LDS 1
#endif
#endif
#ifndef HAVE_ASYNC_LDS
#define HAVE_ASYNC_LDS 0
#endif

__device__ __forceinline__ void wait_async0() {
#if defined(__has_builtin) && __has_builtin(__builtin_amdgcn_s_wait_asynccnt)
  __builtin_amdgcn_s_wait_asynccnt(0);
#else
  asm volatile("s_wait_asynccnt 0" ::: "memory");
#endif
}

#if HAVE_ASYNC_LDS
__device__ __forceinline__ void async_ld_b128(const void* g, void* l) {
  __builtin_amdgcn_global_load_async_to_lds_b128(
      (AS1 i32x4*)(AS1 void*)(void*)g, (AS3 i32x4*)(AS3 void*)l, 0, 0);
}
__device__ __forceinline__ void async_ld_b64(const void* g, void* l) {
  __builtin_amdgcn_global_load_async_to_lds_b64(
      (AS1 i32x2*)(AS1 void*)(void*)g, (AS3 i32x2*)(AS3 void*)l, 0, 0);
}
__device__ __forceinline__ void async_st_b64(void* g, const void* l) {
  __builtin_amdgcn_global_store_async_from_lds_b64(
      (AS1 i32x2*)(AS1 void*)g, (AS3 i32x2*)(AS3 void*)(void*)l, 0, 0);
}
__device__ __forceinline__ void async_st_b32(void* g, const void* l) {
  __builtin_amdgcn_global_store_async_from_lds_b32(
      (AS1 int*)(AS1 void*)g, (AS3 int*)(AS3 void*)(void*)l, 0, 0);
}
#endif

__device__ __forceinline__ v8f wmma_f32_16x16x4(v2f a, v2f b, v8f c) {
  // D = A(16x4) * B(4x16) + C(16x16), fp32, wave32
  return __builtin_amdgcn_wmma_f32_16x16x4_f32(false, a, false, b, (short)0, c,
                                               false, false);
}

// complex 16x16 x 16x16 matmul: C = F * X  (An = -Im F pre-negated since f32
// WMMA has no A/B NEG modifier).
__device__ __forceinline__ void cmm16(const v2f Ar[4], const v2f Ai[4],
                                      const v2f An[4], const v2f Br[4],
                                      const v2f Bi[4], v8f& Cr, v8f& Ci) {
  v8f cr = {}; v8f ci = {};
#pragma unroll
  for (int q = 0; q < 4; ++q) cr = wmma_f32_16x16x4(Ar[q], Br[q], cr);
#pragma unroll
  for (int q = 0; q < 4; ++q) cr = wmma_f32_16x16x4(An[q], Bi[q], cr);
#pragma unroll
  for (int q = 0; q < 4; ++q) ci = wmma_f32_16x16x4(Ar[q], Bi[q], ci);
#pragma unroll
  for (int q = 0; q < 4; ++q) ci = wmma_f32_16x16x4(Ai[q], Br[q], ci);
  Cr = cr; Ci = ci;
}

// One wave performs a 256-point complex FFT in its private LDS region.
// LDS: padded 16x18 float2 matrix, M[a][b] = in[16a+b] at m[a*18+b].
// Output: natural-order bins at m[k], k=0..255.  SGN=-1 fwd, +1 inv.
template <int SGN>
__device__ __forceinline__ void wave_fft256(float2* m) {
  const int lane = threadIdx.x & 31;
  const int lo = lane & 15;        // N index / M index of A
  const int hi = (lane >> 4) & 1;  // K-half selector

  // DFT-16 matrix as WMMA A-operand chunks: F[m][k] = e^{SGN*2pi*i*m*k/16}
  v2f Ar[4], Ai[4], An[4];
#pragma unroll
  for (int q = 0; q < 4; ++q) {
    const int k0 = 4 * q + 2 * hi;
    float s0, c0, s1, c1;
    __sincosf((float)SGN * 2.0f * PI_F * (float)(lo * k0) * (1.0f / 16.0f), &s0, &c0);
    __sincosf((float)SGN * 2.0f * PI_F * (float)(lo * (k0 + 1)) * (1.0f / 16.0f), &s1, &c1);
    Ar[q].x = c0;  Ar[q].y = c1;
    Ai[q].x = s0;  Ai[q].y = s1;
    An[q].x = -s0; An[q].y = -s1;
  }

  v2f Br[4], Bi[4];
  // ---- stage A: U = F * M (DFT over the stride-16 index a) ----
#pragma unroll
  for (int q = 0; q < 4; ++q) {
    const int r0 = 4 * q + 2 * hi;
    const float2 t0 = m[r0 * 18 + lo];
    const float2 t1 = m[(r0 + 1) * 18 + lo];
    Br[q].x = t0.x; Br[q].y = t1.x;
    Bi[q].x = t0.y; Bi[q].y = t1.y;
  }
  v8f cr, ci;
  cmm16(Ar, Ai, An, Br, Bi, cr, ci);

  // twiddle W256^{j*b}, store transposed: T[b][j] = U[j][b]
#pragma unroll
  for (int v = 0; v < 8; ++v) {
    const int mm = v + 8 * hi;  // j
    float sn, cs;
    __sincosf((float)SGN * 2.0f * PI_F * (float)(mm * lo) * (1.0f / 256.0f), &sn, &cs);
    float2 w;
    w.x = cr[v] * cs - ci[v] * sn;
    w.y = cr[v] * sn + ci[v] * cs;
    m[lo * 18 + mm] = w;
  }
  asm volatile("s_wait_dscnt 0" ::: "memory");  // cross-lane LDS RAW, same wave

  // ---- stage B: V = F * T  ->  X[b + 16*j2] = V[j2][b] ----
#pragma unroll
  for (int q = 0; q < 4; ++q) {
    const int r0 = 4 * q + 2 * hi;
    const float2 t0 = m[r0 * 18 + lo];
    const float2 t1 = m[(r0 + 1) * 18 + lo];
    Br[q].x = t0.x; Br[q].y = t1.x;
    Bi[q].x = t0.y; Bi[q].y = t1.y;
  }
  v8f dr, di;
  cmm16(Ar, Ai, An, Br, Bi, dr, di);

#pragma unroll
  for (int v = 0; v < 8; ++v) {
    const int k = lo + 16 * (v + 8 * hi);  // natural-order bin
    float2 w; w.x = dr[v]; w.y = di[v];
    m[k] = w;
  }
  asm volatile("s_wait_dscnt 0" ::: "memory");
}

// ---------------------------------------------------------------------------
// Four-step stage 1: column FFTs (stride 256) + W_65536^{c*k2} (*scale),
// transposed output. FWD: read packed real input z[n]=x[2n]+i x[2n+1], padded.
template <int SGN, bool FWD>
__global__ __launch_bounds__(256) void fft_stage1_kernel(
    const float* __restrict__ xr,    // FWD: per-signal stride 65537
    const float2* __restrict__ zc,   // !FWD: per-signal stride 65536
    float2* __restrict__ dst,        // [s][k2*256 + c]
    float scale) {
  __shared__ float2 sM[8][288];  // 8 waves x (16x18 float2)
  const int s = blockIdx.x >> 5;
  const int c0 = (blockIdx.x & 31) * 8;
  const int tid = threadIdx.x;

  if (FWD) {
    for (int e = tid; e < 2048; e += 256) {
      const int cc = e & 7, r = e >> 3;
      const int n = c0 + cc + 256 * r;
      const float* xs = xr + (size_t)s * 65537u;
      const int i0 = 2 * n;
      float2 z;
      z.x = (i0 < 65537) ? xs[i0] : 0.0f;
      z.y = (i0 + 1 < 65537) ? xs[i0 + 1] : 0.0f;
      sM[cc][(r >> 4) * 18 + (r & 15)] = z;
    }
  } else {
    for (int e = tid; e < 2048; e += 256) {
      const int cc = e & 7, r = e >> 3;
      const int n = c0 + cc + 256 * r;
#if HAVE_ASYNC_LDS
      async_ld_b64(&zc[(size_t)s * 65536u + (size_t)n],
                   &sM[cc][(r >> 4) * 18 + (r & 15)]);
#else
      sM[cc][(r >> 4) * 18 + (r & 15)] = zc[(size_t)s * 65536u + (size_t)n];
#endif
    }
#if HAVE_ASYNC_LDS
    wait_async0();
#endif
  }
  __syncthreads();

  const int wv = tid >> 5;
  wave_fft256<SGN>(&sM[wv][0]);

  // four-step twiddle W_65536^{c*k2}, fold in output scale
  const int lane = tid & 31;
  const int c = c0 + wv;
#pragma unroll
  for (int q = 0; q < 8; ++q) {
    const int k = lane + 32 * q;
    float sn, cs;
    __sincosf((float)SGN * 2.0f * PI_F * (float)(k * c) * (1.0f / 65536.0f), &sn, &cs);
    const float2 u = sM[wv][k];
    float2 w;
    w.x = (u.x * cs - u.y * sn) * scale;
    w.y = (u.x * sn + u.y * cs) * scale;
    sM[wv][k] = w;
  }
  __syncthreads();

  // transposed store (8 consecutive columns -> 64B segments)
  for (int e = tid; e < 2048; e += 256) {
    const int cc = e & 7, k = e >> 3;
    float2* gp = &dst[(size_t)s * 65536u + (size_t)k * 256u + (size_t)(c0 + cc)];
#if HAVE_ASYNC_LDS
    async_st_b64(gp, &sM[cc][k]);  // drained by S_ENDPGM implicit wait-idle
#else
    *gp = sM[cc][k];
#endif
  }
}

// Four-step stage 2: row FFTs, output bin = k2 + 256*k1 (natural order).
// TOCMPLX: write complex spectrum; else write real samples (irfft tail).
template <int SGN, bool TOCMPLX>
__global__ __launch_bounds__(256) void fft_stage2_kernel(
    const float2* __restrict__ src,  // [s][k2*256 + c]
    float2* __restrict__ dstc,       // [s][k2 + 256*k1]
    float* __restrict__ dstr) {      // per-signal stride 65537
  __shared__ float2 sM[8][288];
  const int s = blockIdx.x >> 5;
  const int r0 = (blockIdx.x & 31) * 8;
  const int tid = threadIdx.x;

#if HAVE_ASYNC_LDS
  // 16B async copies; even column pair stays contiguous in the pitch-18 layout
  for (int p = tid; p < 1024; p += 256) {
    const int cpos = (p & 127) * 2;
    const int cc = p >> 7;
    async_ld_b128(&src[(size_t)s * 65536u + (size_t)(r0 + cc) * 256u + (size_t)cpos],
                  &sM[cc][(cpos >> 4) * 18 + (cpos & 15)]);
  }
  wait_async0();
#else
  for (int e = tid; e < 2048; e += 256) {
    const int cpos = e & 255;
    const int cc = e >> 8;
    sM[cc][(cpos >> 4) * 18 + (cpos & 15)] =
        src[(size_t)s * 65536u + (size_t)(r0 + cc) * 256u + (size_t)cpos];
  }
#endif
  __syncthreads();

  const int wv = tid >> 5;
  wave_fft256<SGN>(&sM[wv][0]);
  __syncthreads();

  if (TOCMPLX) {
    for (int e = tid; e < 2048; e += 256) {
      const int cc = e & 7, k1 = e >> 3;
      float2* gp = &dstc[(size_t)s * 65536u + (size_t)k1 * 256u + (size_t)(r0 + cc)];
#if HAVE_ASYNC_LDS
      async_st_b64(gp, &sM[cc][k1]);
#else
      *gp = sM[cc][k1];
#endif
    }
  } else {
    for (int e = tid; e < 2048; e += 256) {
      const int cc = e & 7, k1 = e >> 3;
      const int n = (r0 + cc) + (k1 << 8);
      const size_t base = (size_t)s * 65537u;
      const int i0 = 2 * n;  // y[2n]=Re z', y[2n+1]=Im z', truncate to L
#if HAVE_ASYNC_LDS
      if (i0 + 1 < 65537)  async_st_b64(&dstr[base + i0], &sM[cc][k1]);
      else if (i0 < 65537) async_st_b32(&dstr[base + i0], &sM[cc][k1]);
#else
      if (i0 < 65537)     dstr[base + i0]     = sM[cc][k1].x;
      if (i0 + 1 < 65537) dstr[base + i0 + 1] = sM[cc][k1].y;
#endif
    }
  }
}

// ---------------------------------------------------------------------------
// Spectral processing: unpack rfft bins via conjugate symmetry, apply gain,
// 3-tap magnitude smoothing with phase preservation, repack for inverse FFT.
__device__ __forceinline__ float2 rfft_bin(const float2* __restrict__ Z,
                                           const float* __restrict__ g, int k) {
  const float2 a = Z[k & 65535];
  const float2 b = Z[(65536 - k) & 65535];
  const float er = 0.5f * (a.x + b.x);
  const float ei = 0.5f * (a.y - b.y);
  const float odr = 0.5f * (a.x - b.x);
  const float odi = 0.5f * (a.y + b.y);
  float sn, cs;
  __sincosf(-PI_F * (float)k * (1.0f / 65536.0f), &sn, &cs);
  float2 X;
  X.x = er + (cs * odi + sn * odr);
  X.y = ei + (sn * odi - cs * odr);
  const float gk = g[k];
  X.x *= gk; X.y *= gk;
  return X;
}

__device__ __forceinline__ float bin_mag(const float2* __restrict__ Z,
                                         const float* __restrict__ g, int k) {
  const float2 X = rfft_bin(Z, g, k);
  return sqrtf(X.x * X.x + X.y * X.y);
}

__device__ __forceinline__ float2 y_bin(const float2* __restrict__ Z,
                                        const float* __restrict__ g, int k) {
  const float2 X = rfft_bin(Z, g, k);
  const float m = sqrtf(X.x * X.x + X.y * X.y);
  float ms;
  if (k == 0 || k == 65536)
    ms = m;
  else
    ms = 0.7f * m + 0.15f * bin_mag(Z, g, k - 1) + 0.15f * bin_mag(Z, g, k + 1);
  const float ph = atan2f(X.y, X.x);
  float sn, cs; __sincosf(ph, &sn, &cs);
  float2 Y; Y.x = ms * cs; Y.y = ms * sn;
  return Y;
}

__global__ __launch_bounds__(256) void spectral_kernel(
    const float2* __restrict__ Zall, const float* __restrict__ g,
    float2* __restrict__ Zp) {
  const int s = blockIdx.y;
  const int k = blockIdx.x * 256 + threadIdx.x;  // 0..65535
  const float2* Z = Zall + (size_t)s * 65536u;
  const float2 Ya = y_bin(Z, g, k);
  const float2 Yb = y_bin(Z, g, 65536 - k);
  const float er = 0.5f * (Ya.x + Yb.x);
  const float ei = 0.5f * (Ya.y - Yb.y);
  const float odr = 0.5f * (Ya.x - Yb.x);
  const float odi = 0.5f * (Ya.y + Yb.y);
  float sn, cs; __sincosf(PI_F * (float)k * (1.0f / 65536.0f), &sn, &cs);
  float2 zp;
  zp.x = er - (cs * odi + sn * odr);
  zp.y = ei + (cs * odr - sn * odi);
  Zp[(size_t)s * 65536u + (size_t)k] = zp;
}

// ---------------------------------------------------------------------------
// gain[65537]: band masks * time-mod + spectral enhancement windows + MLP pm.
__global__ __launch_bounds__(256) void gain_kernel(
    const float* __restrict__ bw, const float* __restrict__ fw,
    const float* __restrict__ hp, const float* __restrict__ W1,
    const float* __restrict__ b1, const float* __restrict__ W2,
    const float* __restrict__ b2, const float* __restrict__ W3,
    const float* __restrict__ b3, float* __restrict__ g) {
  __shared__ float red[256];
  __shared__ float shmd;
  const int tid = threadIdx.x;

  if (tid == 0) {
    // mod_depth = MLP(t=0)[1]; tn = 0 exactly so W1 contributes nothing.
    float h1[32], h2[16];
    for (int j = 0; j < 32; ++j) h1[j] = fmaxf(0.0f, b1[j]);
    for (int j = 0; j < 16; ++j) {
      float a = b2[j];
      for (int i = 0; i < 32; ++i) a += h1[i] * W2[i * 16 + j];
      h2[j] = fmaxf(0.0f, a);
    }
    float md = b3[1];
    for (int i = 0; i < 16; ++i) md += h2[i] * W3[i * 8 + 1];
    shmd = md;
  }
  __syncthreads();
  const float md = shmd;

  // pm = mean over [L,4] of hp[idx(t)] * (1 + md*sin(2pi t/L))
  float acc = 0.0f;
  for (int t = tid; t < 65537; t += 256) {
    const int idx = ((int)floorf((float)t * 8.0f / 65537.0f)) & 7;
    const float rs = hp[idx * 4 + 0] + hp[idx * 4 + 1] + hp[idx * 4 + 2] + hp[idx * 4 + 3];
    acc += rs * (1.0f + md * __sinf(2.0f * PI_F * (float)t / 65537.0f));
  }
  red[tid] = acc;
  __syncthreads();
  for (int o = 128; o > 0; o >>= 1) {
    if (tid < o) red[tid] += red[tid + o];
    __syncthreads();
  }
  const float pm = red[0] / (65537.0f * 4.0f);

  const float lows[6]  = {1.0f, 4.0f, 8.0f, 13.0f, 30.0f, 100.0f};
  const float highs[6] = {4.0f, 8.0f, 13.0f, 30.0f, 100.0f, 200.0f};
  for (int k = tid; k < 65537; k += 256) {
    const float f = (float)k * (22050.0f / 131072.0f);
    float gg = 1.0f;
#pragma unroll
    for (int b = 0; b < 6; ++b) {
      const float lo = lows[b], hi = highs[b];
      const float cen = 0.5f * (lo + hi), wid = 0.5f * (hi - lo);
      if (f >= lo && f <= hi) {
        const float z = (f - cen) / (0.5f * wid);
        const float m = __expf(-0.5f * z * z);
        const float tm = __sinf(2.0f * PI_F * cen * (float)k / 22050.0f);
        gg *= 1.0f + m * bw[b] * (1.0f + 0.2f * tm);
      }
    }
    g[k] = gg;
  }
  __syncthreads();

  // spectral enhancement windows (sequential; windows of different harmonics
  // can overlap, e.g. 528*3 == 396*4 == 1584 Hz)
  const float specf[8] = {7.83f, 528.0f, 396.0f, 2.5f, 14.1f, 432.0f, 6.0f, 30.0f};
  for (int j = 0; j < 8; ++j) {
    if (specf[j] >= 11025.0f) continue;
    for (int mi = 1; mi <= 5; ++mi) {
      const float hf = specf[j] * (float)mi;
      if (hf >= 11025.0f) continue;
      const int hidx = (int)roundf(hf * (131072.0f / 22050.0f));
      const int sidx = (hidx - 15 > 0) ? hidx - 15 : 0;
      const int eidx = (hidx + 15 < 65536) ? hidx + 15 : 65536;
      if (sidx >= eidx) continue;
      __syncthreads();
      const int i = sidx + tid;
      if (i <= eidx) {
        const float d = (float)(i - hidx) / 5.0f;  // ws/3 = 5
        const float win = __expf(-0.5f * d * d);
        g[i] *= 1.0f + fw[j] * win * __powf((float)mi, -1.2f) * (1.0f + pm);
      }
    }
  }
}

// ---------------------------------------------------------------------------
extern "C" void kernel_launch(void* const* d_in, const int* in_sizes, int n_in,
                              void* d_out, int out_size, void* d_ws,
                              size_t ws_size, hipStream_t stream) {
  (void)n_in; (void)out_size; (void)ws_size;
  const float* x  = (const float*)d_in[0];
  const float* bw = (const float*)d_in[1];
  const float* fw = (const float*)d_in[2];
  const float* hp = (const float*)d_in[3];
  const float* W1 = (const float*)d_in[4];
  const float* b1 = (const float*)d_in[5];
  const float* W2 = (const float*)d_in[6];
  const float* b2 = (const float*)d_in[7];
  const float* W3 = (const float*)d_in[8];
  const float* b3 = (const float*)d_in[9];
  float* out = (float*)d_out;

  const int S = in_sizes[0] / 65537;  // B*C signals (256)

  float* g = (float*)d_ws;
  float2* bufA = (float2*)((char*)d_ws + (size_t)(1 << 19));
  float2* bufB = bufA + (size_t)S * 65536u;

  gain_kernel<<<1, 256, 0, stream>>>(bw, fw, hp, W1, b1, W2, b2, W3, b3, g);
  fft_stage1_kernel<-1, true ><<<S * 32, 256, 0, stream>>>(x, nullptr, bufA, 1.0f);
  fft_stage2_kernel<-1, true ><<<S * 32, 256, 0, stream>>>(bufA, bufB, nullptr);
  spectral_kernel<<<dim3(256, S), 256, 0, stream>>>(bufB, g, bufA);
  fft_stage1_kernel< 1, false><<<S * 32, 256, 0, stream>>>(nullptr, bufA, bufB,
                                                           1.0f / 65536.0f);
  fft_stage2_kernel< 1, false><<<S * 32, 256, 0, stream>>>(bufB, nullptr, out);
}